// CBF_2130303778824
// MI455X (gfx1250) — compile-verified
//
#include <hip/hip_runtime.h>
#include <cmath>

#define N_STATE 131072
#define HIDDEN  128
#define C_CHUNKS 256
#define CHUNK (N_STATE / C_CHUNKS)   // 512 columns per pass-1 block
#define TILE_ROWS 8
#define TILE_COLS 512
#define NT (HIDDEN / TILE_ROWS)      // 16 k-tiles in pass 2

typedef float v2f __attribute__((ext_vector_type(2)));
typedef float v8f __attribute__((ext_vector_type(8)));
typedef unsigned int v4u __attribute__((ext_vector_type(4)));
typedef int v8i __attribute__((ext_vector_type(8)));
typedef int v4i __attribute__((ext_vector_type(4)));

// ---------------------------------------------------------------------------
// Pass 1: partial GEMV  part[i, c] = sum_{j in chunk c} W0[i,j] * x_norm[j]
// b128 (float4) streaming of W0: 512B per wave per load instruction.
// ---------------------------------------------------------------------------
__global__ __launch_bounds__(256) void cbf_pass1(
    const float* __restrict__ state, const float* __restrict__ xmax,
    const float* __restrict__ xmin,  const float* __restrict__ W0,
    float* __restrict__ wsPart)
{
  __shared__ __align__(16) float sxn[CHUNK];
  const int tid = threadIdx.x;
  const int j0  = blockIdx.x * CHUNK;
  for (int u = tid; u < CHUNK; u += 256) {
    int j = j0 + u;
    float c = 0.5f * (xmax[j] + xmin[j]);
    float r = 0.5f * (xmax[j] - xmin[j]);
    sxn[u] = (state[j] - c) / r;
  }
  __syncthreads();

  const int lane = tid & 31;
  const int wave = tid >> 5;                 // 8 waves, 16 rows each
  const float4* x4 = (const float4*)sxn;
  for (int rr = 0; rr < 16; ++rr) {
    const int i = wave * 16 + rr;
    const float4* w4 = (const float4*)(W0 + (size_t)i * N_STATE + j0);
    float acc = 0.0f;
#pragma unroll
    for (int t = 0; t < CHUNK / 128; ++t) {  // 4 iterations of b128
      float4 w = w4[lane + 32 * t];
      float4 x = x4[lane + 32 * t];
      acc = fmaf(w.x, x.x, acc);
      acc = fmaf(w.y, x.y, acc);
      acc = fmaf(w.z, x.z, acc);
      acc = fmaf(w.w, x.w, acc);
    }
#pragma unroll
    for (int off = 16; off > 0; off >>= 1)   // wave32 butterfly
      acc += __shfl_xor(acc, off, 32);
    if (lane == 0) wsPart[i * C_CHUNKS + blockIdx.x] = acc;
  }
}

// ---------------------------------------------------------------------------
// Tiny MLP forward + backward (single block, 128 threads).
// out[0] = h(state);  wsA = backprop row a with out_row = a @ (W0 / x_range).
// ---------------------------------------------------------------------------
__global__ __launch_bounds__(128) void cbf_mlp(
    const float* __restrict__ wsPart,
    const float* __restrict__ b0,  const float* __restrict__ W1,
    const float* __restrict__ b1,  const float* __restrict__ W2,
    const float* __restrict__ b2,  const float* __restrict__ W3,
    const float* __restrict__ b3,  const float* __restrict__ Wout,
    const float* __restrict__ bout,
    float* __restrict__ wsA, float* __restrict__ out)
{
  __shared__ float sV[HIDDEN], sD1[HIDDEN], sD2[HIDDEN], sD3[HIDDEN], sT[HIDDEN];
  const int i = threadIdx.x;

  float v = b0[i];
  for (int c = 0; c < C_CHUNKS; ++c) v += wsPart[i * C_CHUNKS + c];
  v = tanhf(v);
  sD1[i] = 1.0f - v * v;
  sV[i] = v;
  __syncthreads();

  float v1 = b1[i];
  for (int k = 0; k < HIDDEN; ++k) v1 = fmaf(W1[i * HIDDEN + k], sV[k], v1);
  v1 = tanhf(v1);
  sD2[i] = 1.0f - v1 * v1;
  __syncthreads(); sV[i] = v1; __syncthreads();

  float v2 = b2[i];
  for (int k = 0; k < HIDDEN; ++k) v2 = fmaf(W2[i * HIDDEN + k], sV[k], v2);
  v2 = tanhf(v2);
  sD3[i] = 1.0f - v2 * v2;
  __syncthreads(); sV[i] = v2; __syncthreads();

  float v3 = b3[i];
  for (int k = 0; k < HIDDEN; ++k) v3 = fmaf(W3[i * HIDDEN + k], sV[k], v3);
  __syncthreads(); sV[i] = v3; __syncthreads();

  if (i == 0) {
    float vo = bout[0];
    for (int k = 0; k < HIDDEN; ++k) vo = fmaf(Wout[k], sV[k], vo);
    out[0] = vo;
  }

  float r = 0.0f;
  for (int k = 0; k < HIDDEN; ++k) r = fmaf(Wout[k], W3[k * HIDDEN + i], r);
  r *= sD3[i];
  sT[i] = r; __syncthreads();

  float r3 = 0.0f;
  for (int k = 0; k < HIDDEN; ++k) r3 = fmaf(sT[k], W2[k * HIDDEN + i], r3);
  r3 *= sD2[i];
  __syncthreads(); sT[i] = r3; __syncthreads();

  float r5 = 0.0f;
  for (int k = 0; k < HIDDEN; ++k) r5 = fmaf(sT[k], W1[k * HIDDEN + i], r5);
  wsA[i] = r5 * sD1[i];
}

// ---------------------------------------------------------------------------
// TDM: DMA one 8x512 f32 tile of W0 (rows ktRow..+7, cols jb0..+511) to LDS.
// D# packing per CDNA5 ISA (2-D descriptor,
// higher groups zero). Only wave 0 of the block issues this.
// ---------------------------------------------------------------------------
__device__ __forceinline__ void tdm_load_tile(const float* gtile, unsigned ldsByteOff) {
  unsigned long long ga = (unsigned long long)(uintptr_t)gtile;
  v4u g0;
  g0[0] = 1u;                                   // count=1, is_restore=0, gather=0
  g0[1] = ldsByteOff;                           // lds_addr (bytes, wg-relative)
  g0[2] = (unsigned)(ga & 0xffffffffu);         // global_addr[31:0]
  g0[3] = (unsigned)((ga >> 32) & 0x01ffffffu)  // global_addr[56:32]
          | 0x80000000u;                        // type=2 ("image")
  v8i g1;
  g1[0] = 0x00020000;                           // data_size=2 -> 4 bytes
  g1[1] = 0;                                    // tensor_dim0[15:0]=0 (131072)
  g1[2] = (int)(0x00000002u | (128u << 16));    // dim0[31:16]=2, tensor_dim1=128
  g1[3] = (int)(TILE_COLS << 16);               // tile_dim0 = 512
  g1[4] = TILE_ROWS;                            // tile_dim1 = 8, tile_dim2 = 0
  g1[5] = N_STATE;                              // tensor_dim0_stride[31:0]
  g1[6] = 0;
  g1[7] = 0;
  v4i gz4 = {0, 0, 0, 0};
  v8i gz8 = {0, 0, 0, 0, 0, 0, 0, 0};
  // amdgpu-toolchain (clang-23) 6-arg form: (g0, g1, g2, g3, extra, cpol)
  __builtin_amdgcn_tensor_load_to_lds(g0, g1, gz4, gz4, gz8, 0);
}

// ---------------------------------------------------------------------------
// Pass 2: out[1+j] = (a @ W0)[:,j] * 2/(xmax[j]-xmin[j]).
// TDM double-buffers 8x512 W0 tiles into LDS (TENSORcnt sync); waves build
// B fragments with ds_load and accumulate via V_WMMA_F32_16X16X4_F32.
// A(16x4) = a[k..k+3] replicated to all rows => every D row is the answer.
// ---------------------------------------------------------------------------
__global__ __launch_bounds__(256) void cbf_pass2(
    const float* __restrict__ W0, const float* __restrict__ wsA,
    const float* __restrict__ xmax, const float* __restrict__ xmin,
    float* __restrict__ out)
{
  __shared__ float sa[HIDDEN];
  __shared__ __align__(16) float tile[2][TILE_ROWS * TILE_COLS];  // 2 x 16KB

  const int tid = threadIdx.x;
  if (tid < HIDDEN) sa[tid] = wsA[tid];

  const int lane  = tid & 31;
  const int wave  = tid >> 5;
  const int col   = lane & 15;       // N index within 16-col tile
  const int khalf = lane >> 4;       // lanes 0-15 -> K{0,1}; 16-31 -> K{2,3}
  const size_t jb0 = (size_t)blockIdx.x * TILE_COLS;
  const int wc = wave * 64 + col;    // this wave's first column inside LDS tile

  const unsigned ldsOff0 = (unsigned)(size_t)&tile[0][0];
  const unsigned ldsOff1 = (unsigned)(size_t)&tile[1][0];

  if (wave == 0) tdm_load_tile(W0 + jb0, ldsOff0);   // k-tile 0

  v8f acc0 = {}, acc1 = {}, acc2 = {}, acc3 = {};

  for (int kt = 0; kt < NT; ++kt) {
    if (wave == 0) {
      if (kt + 1 < NT) {
        tdm_load_tile(W0 + (size_t)(kt + 1) * TILE_ROWS * N_STATE + jb0,
                      ((kt + 1) & 1) ? ldsOff1 : ldsOff0);
        __builtin_amdgcn_s_wait_tensorcnt(1);   // tile kt complete
      } else {
        __builtin_amdgcn_s_wait_tensorcnt(0);
      }
    }
    __syncthreads();                            // publish tile kt to all waves

    const float* buf = tile[kt & 1];
#pragma unroll
    for (int kk = 0; kk < TILE_ROWS; kk += 4) { // two K=4 steps per tile
      const int row = kk + 2 * khalf;
      v2f av;
      av.x = sa[kt * TILE_ROWS + row];
      av.y = sa[kt * TILE_ROWS + row + 1];
      const float* bp = buf + row * TILE_COLS + wc;
      v2f b;
      b.x = bp[0];              b.y = bp[TILE_COLS];
      acc0 = __builtin_amdgcn_wmma_f32_16x16x4_f32(false, av, false, b,
                                                   (short)0, acc0, false, false);
      b.x = bp[16];             b.y = bp[TILE_COLS + 16];
      acc1 = __builtin_amdgcn_wmma_f32_16x16x4_f32(false, av, false, b,
                                                   (short)0, acc1, false, false);
      b.x = bp[32];             b.y = bp[TILE_COLS + 32];
      acc2 = __builtin_amdgcn_wmma_f32_16x16x4_f32(false, av, false, b,
                                                   (short)0, acc2, false, false);
      b.x = bp[48];             b.y = bp[TILE_COLS + 48];
      acc3 = __builtin_amdgcn_wmma_f32_16x16x4_f32(false, av, false, b,
                                                   (short)0, acc3, false, false);
    }
    __syncthreads();            // all reads done before tile kt+2 overwrites
  }

  // D row 0: acc[0] of lanes 0-15 (all D rows identical by construction)
  if (lane < 16) {
    float res[4] = {acc0[0], acc1[0], acc2[0], acc3[0]};
#pragma unroll
    for (int t = 0; t < 4; ++t) {
      size_t j = jb0 + (size_t)wave * 64 + (size_t)t * 16 + col;
      float rinv = 2.0f / (xmax[j] - xmin[j]);  // 1 / x_range[j]
      out[1 + j] = res[t] * rinv;
    }
  }
}

// ---------------------------------------------------------------------------
extern "C" void kernel_launch(void* const* d_in, const int* in_sizes, int n_in,
                              void* d_out, int out_size, void* d_ws, size_t ws_size,
                              hipStream_t stream) {
  const float* state = (const float*)d_in[0];
  const float* xmax  = (const float*)d_in[1];
  const float* xmin  = (const float*)d_in[2];
  const float* W0    = (const float*)d_in[3];
  const float* b0    = (const float*)d_in[4];
  const float* W1    = (const float*)d_in[5];
  const float* b1    = (const float*)d_in[6];
  const float* W2    = (const float*)d_in[7];
  const float* b2    = (const float*)d_in[8];
  const float* W3    = (const float*)d_in[9];
  const float* b3    = (const float*)d_in[10];
  const float* Wout  = (const float*)d_in[11];
  const float* bout  = (const float*)d_in[12];

  float* out    = (float*)d_out;
  float* wsPart = (float*)d_ws;                 // 128*C_CHUNKS floats
  float* wsA    = wsPart + HIDDEN * C_CHUNKS;   // 128 floats

  cbf_pass1<<<C_CHUNKS, 256, 0, stream>>>(state, xmax, xmin, W0, wsPart);
  cbf_mlp<<<1, HIDDEN, 0, stream>>>(wsPart, b0, W1, b1, W2, b2, W3, b3,
                                    Wout, bout, wsA, out);
  cbf_pass2<<<N_STATE / TILE_COLS, 256, 0, stream>>>(W0, wsA, xmax, xmin, out);
}